// LuongAttnDecoderRNN_31722628448754
// MI455X (gfx1250) — compile-verified
//
#include <hip/hip_runtime.h>

typedef float v2f __attribute__((ext_vector_type(2)));
typedef float v4f __attribute__((ext_vector_type(4)));
typedef float v8f __attribute__((ext_vector_type(8)));

#define Hd 1024
#define Vv 50257
#define Bb 64
#define Ss 256
#define GPAD 68   // LDS row pitch (floats): 272B = 16*17 -> b128-aligned rows, conflict-free frags

// ---------------------------------------------------------------------------
// Generic fused GEMM:  C[64 x N] = act( A[64 x K](row pitch lda) @ W[N x K]^T + bias )
// 128 threads = 4 waves; each WG does a 64-wide N tile; wave w -> 16 cols.
// Uses V_WMMA_F32_16X16X4_F32 (wave32). W stream staged NT via b128 loads,
// next K-chunk prefetched (global_prefetch_b8) while current chunk computes.
// ---------------------------------------------------------------------------
__global__ __launch_bounds__(128) void k_gemm64(const float* __restrict__ A, int lda,
                                                const float* __restrict__ W,
                                                const float* __restrict__ bias,
                                                float* __restrict__ C,
                                                int K, int N, int act) {
  __shared__ __align__(16) float sA[64 * GPAD];
  __shared__ __align__(16) float sW[64 * GPAD];
  const int tid   = threadIdx.x;
  const int wave  = tid >> 5;
  const int lane  = tid & 31;
  const int lhalf = lane & 15;
  const int hsel  = lane >> 4;     // 0: K pair {0,1}/M rows 0..7 ; 1: K pair {2,3}/M rows 8..15
  const int n0    = blockIdx.x << 6;

  v8f acc[4];
#pragma unroll
  for (int i = 0; i < 4; ++i)
#pragma unroll
    for (int r = 0; r < 8; ++r) acc[i][r] = 0.0f;

  for (int kb = 0; kb < K; kb += 64) {
    // cooperative, coalesced b128 stage of A-chunk [64x64] and W-tile chunk [64x64]
#pragma unroll
    for (int i = tid; i < 1024; i += 128) {
      int r = i >> 4, c = (i & 15) << 2;
      *(v4f*)&sA[r * GPAD + c] = *(const v4f*)(A + (long)r * lda + kb + c);
      int wr = n0 + r; if (wr >= N) wr = N - 1;          // edge tile: clamp (masked at store)
      *(v4f*)&sW[r * GPAD + c] =
          __builtin_nontemporal_load((const v4f*)(W + (long)wr * K + kb + c));
    }
    // prefetch next W K-chunk into cache while this one computes
    const int nkb = kb + 64;
    if (nkb < K) {
      int r = tid >> 1, half = tid & 1;
      int wr = n0 + r; if (wr >= N) wr = N - 1;
      __builtin_prefetch(&W[(long)wr * K + nkb + half * 32], 0, 1);
    }
    __syncthreads();

#pragma unroll 4
    for (int kc = 0; kc < 64; kc += 4) {
      const int col = kc + (hsel << 1);                  // K = kc + {0,1} or kc + {2,3}
      // B fragment: B[k][n] = W[n0 + n][k] ; lane%16 -> n, VGPR -> k within pair
      v2f bf = *(const v2f*)(&sW[(wave * 16 + lhalf) * GPAD + col]);
#pragma unroll
      for (int mt = 0; mt < 4; ++mt) {
        // A fragment: lane%16 -> M row of this 16-row tile
        v2f af = *(const v2f*)(&sA[(mt * 16 + lhalf) * GPAD + col]);
        acc[mt] = __builtin_amdgcn_wmma_f32_16x16x4_f32(
            false, af, false, bf, (short)0, acc[mt], false, false);
      }
    }
    __syncthreads();
  }

  const int v = n0 + wave * 16 + lhalf;                  // output column (N dim)
  if (v < N) {
    const float bv = bias ? bias[v] : 0.0f;
#pragma unroll
    for (int mt = 0; mt < 4; ++mt)
#pragma unroll
      for (int r = 0; r < 8; ++r) {
        int m = mt * 16 + hsel * 8 + r;                  // C/D layout: VGPR r -> M row, +8 hi lanes
        float val = acc[mt][r] + bv;
        if (act == 1)      val = 1.0f / (1.0f + __expf(-val));   // sigmoid
        else if (act == 2) val = tanhf(val);                      // tanh
        C[(long)m * N + v] = val;
      }
  }
}

// ---------------------------------------------------------------------------
// Block reductions (256 threads)
// ---------------------------------------------------------------------------
__device__ __forceinline__ float block_sum(float v, float* sbuf) {
  int tid = threadIdx.x;
  sbuf[tid] = v; __syncthreads();
  for (int s = 128; s > 0; s >>= 1) {
    if (tid < s) sbuf[tid] += sbuf[tid + s];
    __syncthreads();
  }
  float r = sbuf[0]; __syncthreads();
  return r;
}
__device__ __forceinline__ float block_max(float v, float* sbuf) {
  int tid = threadIdx.x;
  sbuf[tid] = v; __syncthreads();
  for (int s = 128; s > 0; s >>= 1) {
    if (tid < s) sbuf[tid] = fmaxf(sbuf[tid], sbuf[tid + s]);
    __syncthreads();
  }
  float r = sbuf[0]; __syncthreads();
  return r;
}

// ---------------------------------------------------------------------------
// Prep: X[:,0:3072] = [embedding(input_step) | h0 | input_context]; emo gather
// ---------------------------------------------------------------------------
__global__ __launch_bounds__(256) void k_prep(const int* __restrict__ step,
                                              const int* __restrict__ stepE,
                                              const float* __restrict__ h0,
                                              const float* __restrict__ ctx,
                                              const float* __restrict__ emb,
                                              const float* __restrict__ embE,
                                              float* __restrict__ X,
                                              float* __restrict__ emo) {
  int t = blockIdx.x * 256 + threadIdx.x;      // < 64*3072
  int b = t / 3072, j = t - b * 3072;
  float v;
  if (j < Hd) {
    v = emb[(long)step[b] * Hd + j];
    emo[b * Hd + j] = embE[(long)stepE[b] * Hd + j];
  } else if (j < 2 * Hd) {
    v = h0[b * Hd + (j - Hd)];
  } else {
    v = ctx[b * Hd + (j - 2 * Hd)];
  }
  X[(long)b * 4096 + j] = v;
}

// emo_g = emo * sigmoid(M_read);  also X[:,3072:4096] = emo_g
__global__ __launch_bounds__(256) void k_emogate(const float* __restrict__ emo,
                                                 const float* __restrict__ gate,
                                                 float* __restrict__ emo_g,
                                                 float* __restrict__ X) {
  int t = blockIdx.x * 256 + threadIdx.x;      // < 65536
  int b = t >> 10, j = t & 1023;
  float eg = emo[t] * gate[t];
  emo_g[t] = eg;
  X[(long)b * 4096 + 3072 + j] = eg;
}

// GRU elementwise: r,z,n -> h_new (PyTorch gate order r,z,n)
__global__ __launch_bounds__(256) void k_gru(const float* __restrict__ gi,
                                             const float* __restrict__ gh,
                                             const float* __restrict__ h0,
                                             float* __restrict__ Y,
                                             float* __restrict__ out_hidden) {
  int t = blockIdx.x * 256 + threadIdx.x;      // < 65536
  int b = t >> 10, j = t & 1023;
  const float* Gi = gi + (long)b * 3072;
  const float* Gh = gh + (long)b * 3072;
  float r = 1.0f / (1.0f + __expf(-(Gi[j] + Gh[j])));
  float z = 1.0f / (1.0f + __expf(-(Gi[Hd + j] + Gh[Hd + j])));
  float n = tanhf(Gi[2 * Hd + j] + r * Gh[2 * Hd + j]);
  float h = (1.0f - z) * n + z * h0[t];
  Y[(long)b * 2048 + j] = h;                   // left half of [h_new | context]
  out_hidden[t] = h;
}

// new_M_emo = emo_g * sigmoid(M_write)
__global__ __launch_bounds__(256) void k_writegate(const float* __restrict__ emo_g,
                                                   const float* __restrict__ gate,
                                                   float* __restrict__ out_newM) {
  int t = blockIdx.x * 256 + threadIdx.x;
  out_newM[t] = emo_g[t] * gate[t];
}

// scores[b][s] = sum_h h_new[b][h] * enc[s][b][h]
__global__ __launch_bounds__(256) void k_scores(const float* __restrict__ Y,
                                                const float* __restrict__ enc,
                                                float* __restrict__ scores) {
  __shared__ float sbuf[256];
  int b = blockIdx.x >> 8;
  int s = blockIdx.x & 255;
  int tid = threadIdx.x;
  const float* e = enc + (long)s * (Bb * Hd) + (long)b * Hd;
  const float* h = Y + (long)b * 2048;
  float acc = 0.0f;
  for (int j = tid; j < Hd; j += 256) acc += h[j] * e[j];
  acc = block_sum(acc, sbuf);
  if (tid == 0) scores[b * Ss + s] = acc;
}

// softmax over S=256 per batch row (in place)
__global__ __launch_bounds__(256) void k_softmaxS(float* __restrict__ attn) {
  __shared__ float sbuf[256];
  int b = blockIdx.x, tid = threadIdx.x;
  float v = attn[b * Ss + tid];
  float m = block_max(v, sbuf);
  float e = __expf(v - m);
  float s = block_sum(e, sbuf);
  attn[b * Ss + tid] = e / s;
}

// context[b][h] = sum_s attn[b][s] * enc[s][b][h]; also Y[:,1024:2048] = context
__global__ __launch_bounds__(256) void k_context(const float* __restrict__ attn,
                                                 const float* __restrict__ enc,
                                                 float* __restrict__ Y,
                                                 float* __restrict__ out_ctx) {
  int t = blockIdx.x * 256 + threadIdx.x;      // < 65536
  int b = t >> 10, h = t & 1023;
  const float* aw = attn + b * Ss;
  float acc = 0.0f;
  for (int s = 0; s < Ss; ++s)
    acc += aw[s] * enc[(long)s * (Bb * Hd) + (long)b * Hd + h];
  Y[(long)b * 2048 + Hd + h] = acc;
  out_ctx[t] = acc;
}

// Per batch row: alpha-gate dot product + per-group (ememory) max & expsum
__global__ __launch_bounds__(256) void k_stats(const float* __restrict__ logits,
                                               const int* __restrict__ emem,
                                               const float* __restrict__ Wa,
                                               const float* __restrict__ ba,
                                               float* __restrict__ stats) {
  __shared__ float sbuf[256];
  int b = blockIdx.x, tid = threadIdx.x;
  const float* L = logits + (long)b * Vv;
  float dot = 0.0f, mE = -3.0e38f, mG = -3.0e38f;
  for (int v = tid; v < Vv; v += 256) {
    float l = L[v];
    dot += l * Wa[v];
    if (emem[v] == 1) mE = fmaxf(mE, l);
    else              mG = fmaxf(mG, l);
  }
  dot = block_sum(dot, sbuf);
  mE  = block_max(mE, sbuf);
  mG  = block_max(mG, sbuf);
  float sE = 0.0f, sG = 0.0f;
  for (int v = tid; v < Vv; v += 256) {
    float l = L[v];
    if (emem[v] == 1) sE += __expf(l - mE);
    else              sG += __expf(l - mG);
  }
  sE = block_sum(sE, sbuf);
  sG = block_sum(sG, sbuf);
  if (tid == 0) {
    float g = 1.0f / (1.0f + __expf(-(dot + ba[0])));
    float* st = stats + b * 8;
    st[0] = g; st[1] = mE; st[2] = sE; st[3] = mG; st[4] = sG;
  }
}

// output[b][v] = mask ? softmaxE * g : softmaxG * (1-g)  (NT store: write-once)
__global__ __launch_bounds__(256) void k_final(const float* __restrict__ logits,
                                               const int* __restrict__ emem,
                                               const float* __restrict__ stats,
                                               float* __restrict__ out) {
  long t = (long)blockIdx.x * 256 + threadIdx.x;
  if (t >= (long)Bb * Vv) return;
  int b = (int)(t / Vv);
  int v = (int)(t - (long)b * Vv);
  const float* st = stats + b * 8;
  float l = logits[t];
  float o;
  if (emem[v] == 1) o = __expf(l - st[1]) / st[2] * st[0];
  else              o = __expf(l - st[3]) / st[4] * (1.0f - st[0]);
  __builtin_nontemporal_store(o, &out[t]);
}

// ---------------------------------------------------------------------------
extern "C" void kernel_launch(void* const* d_in, const int* in_sizes, int n_in,
                              void* d_out, int out_size, void* d_ws, size_t ws_size,
                              hipStream_t stream) {
  const int*   input_step = (const int*)d_in[0];
  const int*   input_emot = (const int*)d_in[1];
  const float* last_hidden = (const float*)d_in[2];   // [1,B,H] -> h0
  const float* input_ctx   = (const float*)d_in[3];
  const float* enc         = (const float*)d_in[4];
  const int*   ememory     = (const int*)d_in[5];
  const float* embedding   = (const float*)d_in[6];
  const float* emo_emb     = (const float*)d_in[7];
  const float* W_read  = (const float*)d_in[8];
  const float* b_read  = (const float*)d_in[9];
  const float* W_write = (const float*)d_in[10];
  const float* b_write = (const float*)d_in[11];
  const float* W_ih    = (const float*)d_in[12];
  const float* W_hh    = (const float*)d_in[13];
  const float* b_ih    = (const float*)d_in[14];
  const float* b_hh    = (const float*)d_in[15];
  const float* W_concat = (const float*)d_in[16];
  const float* b_concat = (const float*)d_in[17];
  const float* W_out   = (const float*)d_in[18];
  const float* b_out   = (const float*)d_in[19];
  const float* W_alpha = (const float*)d_in[20];
  const float* b_alpha = (const float*)d_in[21];

  float* out = (float*)d_out;
  // d_out layout: output[64*50257] | hidden[64*1024] | new_M_emo[64*1024] | context[64*1024]
  float* out_output = out;
  float* out_hidden = out + (long)Bb * Vv;
  float* out_newM   = out_hidden + Bb * Hd;
  float* out_ctx    = out_newM + Bb * Hd;

  float* ws = (float*)d_ws;
  float* X        = ws;                       // [64][4096]  = [lw | h0 | ctx | emo_g]
  float* emo      = X + 64 * 4096;            // [64][1024]
  float* emo_g    = emo + 64 * Hd;            // [64][1024]
  float* tmpgate  = emo_g + 64 * Hd;          // [64][1024]  (M_read / M_write sigmoid)
  float* gi       = tmpgate + 64 * Hd;        // [64][3072]
  float* gh       = gi + 64 * 3072;           // [64][3072]
  float* Y        = gh + 64 * 3072;           // [64][2048]  = [h_new | context]
  float* concat_o = Y + 64 * 2048;            // [64][1024]
  float* attn     = concat_o + 64 * Hd;       // [64][256]
  float* logits   = attn + 64 * Ss;           // [64][50257]
  float* stats    = logits + (long)Bb * Vv;   // [64][8]

  // 1) gathers + input concat
  k_prep<<<(64 * 3072) / 256, 256, 0, stream>>>(input_step, input_emot, last_hidden,
                                                input_ctx, embedding, emo_emb, X, emo);
  // 2) M_read = sigmoid(X[:,:3072] @ W_read^T + b_read)
  k_gemm64<<<Hd / 64, 128, 0, stream>>>(X, 4096, W_read, b_read, tmpgate, 3072, Hd, 1);
  // 3) emo_g
  k_emogate<<<(64 * Hd) / 256, 256, 0, stream>>>(emo, tmpgate, emo_g, X);
  // 4) gi = X @ W_ih^T + b_ih ; gh = h0 @ W_hh^T + b_hh
  k_gemm64<<<3072 / 64, 128, 0, stream>>>(X, 4096, W_ih, b_ih, gi, 4096, 3072, 0);
  k_gemm64<<<3072 / 64, 128, 0, stream>>>(last_hidden, Hd, W_hh, b_hh, gh, Hd, 3072, 0);
  // 5) GRU elementwise -> h_new (into Y left half + d_out hidden)
  k_gru<<<(64 * Hd) / 256, 256, 0, stream>>>(gi, gh, last_hidden, Y, out_hidden);
  // 6) M_write = sigmoid(h_new @ W_write^T + b_write); new_M_emo = emo_g * M_write
  k_gemm64<<<Hd / 64, 128, 0, stream>>>(Y, 2048, W_write, b_write, tmpgate, Hd, Hd, 1);
  k_writegate<<<(64 * Hd) / 256, 256, 0, stream>>>(emo_g, tmpgate, out_newM);
  // 7) Luong dot attention
  k_scores<<<Bb * Ss, 256, 0, stream>>>(Y, enc, attn);
  k_softmaxS<<<Bb, 256, 0, stream>>>(attn);
  k_context<<<(64 * Hd) / 256, 256, 0, stream>>>(attn, enc, Y, out_ctx);
  // 8) concat_out = tanh([h_new|context] @ W_concat^T + b_concat)
  k_gemm64<<<Hd / 64, 128, 0, stream>>>(Y, 2048, W_concat, b_concat, concat_o, 2048, Hd, 2);
  // 9) big GEMM: logits = concat_out @ W_out^T + b_out   (206 MB weight stream)
  k_gemm64<<<(Vv + 63) / 64, 128, 0, stream>>>(concat_o, Hd, W_out, b_out, logits, Hd, Vv, 0);
  // 10) gate + dual-group softmax stats, then final gated output
  k_stats<<<Bb, 256, 0, stream>>>(logits, ememory, W_alpha, b_alpha, stats);
  k_final<<<(int)(((long)Bb * Vv + 255) / 256), 256, 0, stream>>>(logits, ememory, stats,
                                                                  out_output);
}